// inv_71725953843376
// MI455X (gfx1250) — compile-verified
//
#include <hip/hip_runtime.h>

// Involution: B=4, C=256, G=16, K=7, S=1, P=3, R=4, H=W=56
#define Bsz 4
#define Cch 256
#define Gg  16
#define Kdim 7
#define Pp  3
#define Hh  56
#define Ww  56
#define HW  (Hh * Ww)          // 3136
#define Npix (Bsz * HW)        // 12544
#define CR  64                 // C / R
#define KKG (Kdim * Kdim * Gg) // 784
#define KK  (Kdim * Kdim)      // 49
#define CPG (Cch / Gg)         // 16
#define TS  4                  // spatial tile side (4x4 = 16 pixels = WMMA N)
#define PS  (TS + Kdim - 1)    // 10: patch side
#define PSQ (PS * PS)          // 100 elements per channel
#define KERN_FLOATS (KKG * 16) // 12544 floats = 50176 B
#define PATCH_FLOATS (Cch * PSQ) // 25600 floats = 102400 B
#define LDS_BYTES ((KERN_FLOATS + PATCH_FLOATS) * 4) // 152576 B (< 320 KB/WGP)

typedef __attribute__((ext_vector_type(2))) float v2f;
typedef __attribute__((ext_vector_type(8))) float v8f;

// ---------------------------------------------------------------------------
// Kernel 1: h[64][12544] = reduce_w[64][256] @ x(view [256][12544]) + reduce_b
// One wave per (M-tile, N-tile): 4 * 784 = 3136 waves. FP32 WMMA 16x16x4.
// ---------------------------------------------------------------------------
__global__ __launch_bounds__(256) void inv_gemm_reduce(
    const float* __restrict__ x, const float* __restrict__ rw,
    const float* __restrict__ rb, float* __restrict__ h) {
  const int lane = threadIdx.x & 31;
  const int gw   = blockIdx.x * (blockDim.x >> 5) + (threadIdx.x >> 5);
  const int tile = gw >> 2;              // 0..783 N-tile (16 raster pixels)
  const int mt   = gw & 3;               // 0..3   M-tile
  const int m0   = mt * 16;
  const int n0   = tile * 16;
  const int b    = n0 / HW;              // whole tile inside one image
  const int sp0  = n0 - b * HW;
  const int lane_lo = lane & 15;
  const int krow    = (lane >> 4) << 1;  // 0 or 2: K-pair selector

  const float* xb = x + (size_t)b * Cch * HW + sp0 + lane_lo;
  const float* wa = rw + (size_t)(m0 + lane_lo) * Cch + krow;

  v8f acc = {};
#pragma unroll 4
  for (int k0 = 0; k0 < Cch; k0 += 4) {
    v2f a = *(const v2f*)(wa + k0);              // A: reduce_w rows, K pair
    const float* xp = xb + (size_t)(k0 + krow) * HW;
    v2f bf;                                      // B: 16 coalesced pixels
    bf.x = xp[0];
    bf.y = xp[HW];
    acc = __builtin_amdgcn_wmma_f32_16x16x4_f32(
        false, a, false, bf, (short)0, acc, false, false);
  }

  const int col  = n0 + lane_lo;
  const int mrow = m0 + ((lane >> 4) << 3);
#pragma unroll
  for (int r = 0; r < 8; ++r)
    h[(size_t)(mrow + r) * Npix + col] = acc[r] + rb[mrow + r];
}

// ---------------------------------------------------------------------------
// Kernel 2 (fused), one 256-thread block per 4x4 spatial tile (784 tiles):
//  0) issue async global->LDS staging of the 10x10x256 x-patch (zero-pad OOB)
//  1) Phase A (overlapped with 0): kern[784][16] = span_w @ h_tile + b -> LDS
//  2) s_wait_asynccnt 0, barrier
//  3) Phase B: branch-free involution multiply-reduce entirely out of LDS
// ---------------------------------------------------------------------------
__global__ __launch_bounds__(256) void inv_span_apply(
    const float* __restrict__ x, const float* __restrict__ sw,
    const float* __restrict__ sb, const float* __restrict__ h,
    float* __restrict__ out) {
  extern __shared__ float smem[];
  float* kern  = smem;                 // [784][16], LDS offset 0
  float* patch = smem + KERN_FLOATS;   // [256][10][10], LDS offset 50176 B

  const int tid  = threadIdx.x;
  const int lane = tid & 31;
  const int wv   = tid >> 5;
  const int blk  = blockIdx.x;
  const int b    = blk / (14 * 14);
  const int r2   = blk - b * (14 * 14);
  const int y0   = (r2 / 14) * TS;
  const int x0   = (r2 - (r2 / 14) * 14) * TS;
  const int lane_lo = lane & 15;
  const int krow    = (lane >> 4) << 1;

  // ---- Stage 0: async global->LDS patch staging (per-lane addresses) ----
  {
    const float* xb = x + (size_t)b * Cch * HW;
    const int gyb = y0 - Pp, gxb = x0 - Pp;
    for (int i = 0; i < PSQ; ++i) {           // 100 iters * 256 threads
      const int e   = i * 256 + tid;          // element in [256][10][10]
      const int c   = e / PSQ;
      const int rem = e - c * PSQ;
      const int ry  = rem / PS;
      const int rx  = rem - ry * PS;
      const int gy  = gyb + ry;
      const int gx  = gxb + rx;
      if ((unsigned)gy < (unsigned)Hh && (unsigned)gx < (unsigned)Ww) {
        const float* gp = xb + (size_t)c * HW + gy * Ww + gx;
        const unsigned lds = (unsigned)((KERN_FLOATS + e) << 2); // byte addr
        asm volatile("global_load_async_to_lds_b32 %0, %1, off"
                     :: "v"(lds), "v"(gp) : "memory");
      } else {
        patch[e] = 0.0f;                      // zero padding via ds_store
      }
    }
  }

  // ---- Phase A: FP32 WMMA GEMM, span_w[784][64] @ h_tile[64][16] ----
  // lane column: pixel p = lane_lo -> (py,px) = (p>>2, p&3)
  const int coln = b * HW + (y0 + (lane_lo >> 2)) * Ww + x0 + (lane_lo & 3);
  const float* hcol = h + coln;
  for (int mt = wv; mt < 49; mt += 8) {
    const int m0 = mt * 16;
    const float* wa = sw + (size_t)(m0 + lane_lo) * CR + krow;
    v8f acc = {};
#pragma unroll
    for (int k0 = 0; k0 < CR; k0 += 4) {
      v2f a = *(const v2f*)(wa + k0);
      const float* hp = hcol + (size_t)(k0 + krow) * Npix;
      v2f bf;
      bf.x = hp[0];
      bf.y = hp[Npix];
      acc = __builtin_amdgcn_wmma_f32_16x16x4_f32(
          false, a, false, bf, (short)0, acc, false, false);
    }
    const int mrow = m0 + ((lane >> 4) << 3);
#pragma unroll
    for (int r = 0; r < 8; ++r)
      kern[(mrow + r) * 16 + lane_lo] = acc[r] + sb[mrow + r];
  }

  // ---- Wait for async staging, then make LDS visible to all waves ----
  asm volatile("s_wait_asynccnt 0x0" ::: "memory");
  __syncthreads();

  // ---- Phase B: branch-free involution out of LDS ----
  const int p  = tid & 15;                 // pixel in tile
  const int g  = tid >> 4;                 // group
  const int py = p >> 2, px = p & 3;

  float kv[KK];
#pragma unroll
  for (int kk = 0; kk < KK; ++kk)
    kv[kk] = kern[(g * KK + kk) * 16 + p];

  const float* pbase = patch + (g * CPG) * PSQ + py * PS + px;
  float* op = out + ((size_t)b * Cch + g * CPG) * HW + (y0 + py) * Ww + (x0 + px);

#pragma unroll 4
  for (int c = 0; c < CPG; ++c) {
    const float* pp = pbase + c * PSQ;
    float a = 0.0f;
#pragma unroll
    for (int dy = 0; dy < Kdim; ++dy)
#pragma unroll
      for (int dx = 0; dx < Kdim; ++dx)
        a += kv[dy * Kdim + dx] * pp[dy * PS + dx];
    op[c * HW] = a;
  }
}

// ---------------------------------------------------------------------------
extern "C" void kernel_launch(void* const* d_in, const int* in_sizes, int n_in,
                              void* d_out, int out_size, void* d_ws, size_t ws_size,
                              hipStream_t stream) {
  const float* x        = (const float*)d_in[0];
  const float* reduce_w = (const float*)d_in[1];
  const float* reduce_b = (const float*)d_in[2];
  const float* span_w   = (const float*)d_in[3];
  const float* span_b   = (const float*)d_in[4];
  float* out  = (float*)d_out;
  float* h_ws = (float*)d_ws;            // 64 * 12544 floats = 3.2 MB

  inv_gemm_reduce<<<dim3(392), dim3(256), 0, stream>>>(x, reduce_w, reduce_b, h_ws);
  inv_span_apply<<<dim3(784), dim3(256), LDS_BYTES, stream>>>(x, span_w, span_b, h_ws, out);
}